// Transformer_75900662055205
// MI455X (gfx1250) — compile-verified
//
#include <hip/hip_runtime.h>
#include <hip/hip_bf16.h>

typedef float v2f __attribute__((ext_vector_type(2)));
typedef float v8f __attribute__((ext_vector_type(8)));

constexpr int kB  = 2;
constexpr int kC  = 256;
constexpr int kS  = 2304;                        // 48*48
constexpr int kH  = 8;
constexpr int kDH = 32;
constexpr size_t kNE = (size_t)kB * kS * kC;     // 1,179,648
constexpr int kPW  = 50;                         // padded width
constexpr int kPP  = kPW * kPW;                  // 2500 per plane
constexpr size_t kNP = (size_t)kB * kC * kPP;    // 1,280,000 per padded tensor

static __device__ __forceinline__ v8f wmma_f32(v2f a, v2f b, v8f c) {
  // D = A(16x4) * B(4x16) + C(16x16), fp32 in / fp32 accumulate
  return __builtin_amdgcn_wmma_f32_16x16x4_f32(false, a, false, b, (short)0, c,
                                               false, false);
}

// ---------------------------------------------------------------------------
__global__ void zero_kernel(float* __restrict__ p) {
  size_t idx = (size_t)blockIdx.x * blockDim.x + threadIdx.x;
  p[idx] = 0.0f;
}

// ---------------------------------------------------------------------------
// [b,C,S] -> [b,S,C] for both x and ref
__global__ void transpose_kernel(const float* __restrict__ x,
                                 const float* __restrict__ r,
                                 float* __restrict__ content,
                                 float* __restrict__ refer) {
  size_t idx = (size_t)blockIdx.x * blockDim.x + threadIdx.x;  // < kNE
  int c  = (int)(idx & 255u);
  int bs = (int)(idx >> 8);            // b*S + s
  int b  = bs / kS;
  int s  = bs - b * kS;
  size_t src = ((size_t)(b * kC + c)) * kS + s;
  content[idx] = x[src];
  refer[idx]   = r[src];
}

// ---------------------------------------------------------------------------
// out[M,N] = act(A[M,K] @ W[K,N] + bias[N]); one wave per 16x64 tile (4 accs).
__global__ void gemm_bias_act_kernel(const float* __restrict__ A,
                                     const float* __restrict__ W,
                                     const float* __restrict__ bias,
                                     float* __restrict__ out,
                                     int K, int N, int do_relu) {
  int wid  = blockIdx.x * 8 + (threadIdx.x >> 5);
  int lane = threadIdx.x & 31;
  int tg = wid & 3;                    // N-group of 4 tiles (N == 256)
  int tm = wid >> 2;
  int half = lane >> 4, lm = lane & 15;
  const float* Arow = A + (size_t)(tm * 16 + lm) * K;
  int col0 = tg * 64 + lm;
  v8f a0 = {}, a1 = {}, a2 = {}, a3 = {};
  for (int k0 = 0; k0 < K; k0 += 4) {
    int ka = k0 + 2 * half;
    v2f a;  a.x = Arow[ka]; a.y = Arow[ka + 1];
    const float* W0 = W + (size_t)ka * N + col0;
    const float* W1 = W + (size_t)(ka + 1) * N + col0;
    v2f b0; b0.x = W0[0];  b0.y = W1[0];
    v2f b1; b1.x = W0[16]; b1.y = W1[16];
    v2f b2; b2.x = W0[32]; b2.y = W1[32];
    v2f b3; b3.x = W0[48]; b3.y = W1[48];
    a0 = wmma_f32(a, b0, a0);
    a1 = wmma_f32(a, b1, a1);
    a2 = wmma_f32(a, b2, a2);
    a3 = wmma_f32(a, b3, a3);
  }
  v8f accs[4] = {a0, a1, a2, a3};
#pragma unroll
  for (int g = 0; g < 4; ++g) {
    int col = col0 + g * 16;
    float bv = bias[col];
#pragma unroll
    for (int rr = 0; rr < 8; ++rr) {
      float v = accs[g][rr] + bv;
      if (do_relu) v = fmaxf(v, 0.0f);
      out[(size_t)(tm * 16 + rr + 8 * half) * N + col] = v;
    }
  }
}

// ---------------------------------------------------------------------------
// scores[b,h,i,j] = (q . k) / sqrt(32); one wave per 16x64 tile; into d_out.
__global__ void scores_kernel(const float* __restrict__ q,
                              const float* __restrict__ k,
                              float* __restrict__ scores) {
  int wid  = blockIdx.x * 8 + (threadIdx.x >> 5);
  int lane = threadIdx.x & 31;
  int tg  = wid % 36;                  // group of 4 j-tiles (144/4)
  int rem = wid / 36;
  int ti  = rem % 144;
  int bh  = rem / 144;
  int b = bh >> 3, h = bh & 7;
  int half = lane >> 4, lm = lane & 15;
  const float* qrow = q + ((size_t)(b * kS + ti * 16 + lm)) * kC + h * kDH;
  int j0 = tg * 64 + lm;
  const float* k0p = k + ((size_t)(b * kS + j0 +  0)) * kC + h * kDH;
  const float* k1p = k + ((size_t)(b * kS + j0 + 16)) * kC + h * kDH;
  const float* k2p = k + ((size_t)(b * kS + j0 + 32)) * kC + h * kDH;
  const float* k3p = k + ((size_t)(b * kS + j0 + 48)) * kC + h * kDH;
  v8f a0 = {}, a1 = {}, a2 = {}, a3 = {};
#pragma unroll
  for (int kk = 0; kk < kDH; kk += 4) {
    int ka = kk + 2 * half;
    v2f a;  a.x  = qrow[ka]; a.y  = qrow[ka + 1];
    v2f b0; b0.x = k0p[ka];  b0.y = k0p[ka + 1];
    v2f b1; b1.x = k1p[ka];  b1.y = k1p[ka + 1];
    v2f b2; b2.x = k2p[ka];  b2.y = k2p[ka + 1];
    v2f b3; b3.x = k3p[ka];  b3.y = k3p[ka + 1];
    a0 = wmma_f32(a, b0, a0);
    a1 = wmma_f32(a, b1, a1);
    a2 = wmma_f32(a, b2, a2);
    a3 = wmma_f32(a, b3, a3);
  }
  const float scale = 0.17677669529663687f;  // 1/sqrt(32)
  v8f accs[4] = {a0, a1, a2, a3};
  size_t base = ((size_t)(b * kH + h) * kS + ti * 16) * kS + tg * 64 + lm;
#pragma unroll
  for (int g = 0; g < 4; ++g)
#pragma unroll
    for (int rr = 0; rr < 8; ++rr)
      scores[base + (size_t)(rr + 8 * half) * kS + g * 16] = accs[g][rr] * scale;
}

// ---------------------------------------------------------------------------
// Top-1 over head axis: keep entries equal to the head-max, else -1e18.
__global__ void headmask_kernel(float* __restrict__ p) {
  size_t idx = (size_t)blockIdx.x * blockDim.x + threadIdx.x;  // < kB*S*S
  const size_t SS = (size_t)kS * kS;
  size_t b = idx / SS;
  size_t r = idx - b * SS;
  float* base = p + b * kH * SS + r;
  float m = base[0];
#pragma unroll
  for (int h = 1; h < kH; ++h) m = fmaxf(m, base[(size_t)h * SS]);
#pragma unroll
  for (int h = 0; h < kH; ++h) {
    float v = base[(size_t)h * SS];
    base[(size_t)h * SS] = (v < m) ? -1e18f : v;
  }
}

// ---------------------------------------------------------------------------
// In-place row softmax over j; one 256-thread block per (b,h,i) row.
__global__ void softmax_kernel(float* __restrict__ probs) {
  __shared__ float red[256];
  float* row = probs + (size_t)blockIdx.x * kS;
  int tid = threadIdx.x;
  float v[9];
  float m = -3.4e38f;
#pragma unroll
  for (int t = 0; t < 9; ++t) { v[t] = row[t * 256 + tid]; m = fmaxf(m, v[t]); }
  red[tid] = m; __syncthreads();
  for (int s = 128; s > 0; s >>= 1) {
    if (tid < s) red[tid] = fmaxf(red[tid], red[tid + s]);
    __syncthreads();
  }
  m = red[0]; __syncthreads();
  float sum = 0.0f;
#pragma unroll
  for (int t = 0; t < 9; ++t) { v[t] = expf(v[t] - m); sum += v[t]; }
  red[tid] = sum; __syncthreads();
  for (int s = 128; s > 0; s >>= 1) {
    if (tid < s) red[tid] += red[tid + s];
    __syncthreads();
  }
  float inv = 1.0f / red[0];
#pragma unroll
  for (int t = 0; t < 9; ++t) row[t * 256 + tid] = v[t] * inv;
}

// ---------------------------------------------------------------------------
// ctx[b,h,i,d] = probs @ v; one wave per 16x32 tile (full head dim).
// Result written into the INTERIOR of the padded conv1 input [B,C,50,50].
__global__ void ctx_kernel(const float* __restrict__ p,
                           const float* __restrict__ v,
                           float* __restrict__ pad1) {
  int wid  = blockIdx.x * 8 + (threadIdx.x >> 5);
  int lane = threadIdx.x & 31;
  int ti = wid % 144;
  int bh = wid / 144;
  int b = bh >> 3, h = bh & 7;
  int half = lane >> 4, lm = lane & 15;
  const float* Arow = p + ((size_t)(b * kH + h) * kS + ti * 16 + lm) * kS;
  const float* V0 = v + (size_t)b * kS * kC + h * kDH + lm;
  const float* V1 = V0 + 16;
  v8f a0 = {}, a1 = {};
  for (int k0 = 0; k0 < kS; k0 += 4) {
    if ((k0 & 63) == 0) __builtin_prefetch(Arow + k0 + 128, 0, 3);
    int ka = k0 + 2 * half;
    v2f a;  a.x  = Arow[ka];              a.y  = Arow[ka + 1];
    v2f b0; b0.x = V0[(size_t)ka * kC];   b0.y = V0[(size_t)(ka + 1) * kC];
    v2f b1; b1.x = V1[(size_t)ka * kC];   b1.y = V1[(size_t)(ka + 1) * kC];
    a0 = wmma_f32(a, b0, a0);
    a1 = wmma_f32(a, b1, a1);
  }
  v8f accs[2] = {a0, a1};
#pragma unroll
  for (int g = 0; g < 2; ++g) {
    int ch = h * kDH + g * 16 + lm;
    size_t plane = ((size_t)(b * kC + ch)) * kPP;
#pragma unroll
    for (int rr = 0; rr < 8; ++rr) {
      int s = ti * 16 + rr + 8 * half;
      int y = s / 48, x = s - y * 48;
      pad1[plane + (size_t)(y + 1) * kPW + (x + 1)] = accs[g][rr];
    }
  }
}

// ---------------------------------------------------------------------------
// 3x3 SAME conv as 9 shifted WMMA GEMMs over padded input [B,C,50,50].
// M = co (16/tile), N = 64 pixels/wave, K = ci, taps accumulated in regs.
// mode 1: SELU epilogue, write padded interior (outPad). mode 0: write flat.
__global__ void conv3x3_wmma_kernel(const float* __restrict__ pad,
                                    const float* __restrict__ w,   // [CO,CI,3,3]
                                    const float* __restrict__ bias,
                                    float* __restrict__ outPad,
                                    float* __restrict__ outFlat, int mode) {
  int wid  = blockIdx.x * 8 + (threadIdx.x >> 5);
  int lane = threadIdx.x & 31;
  int tn  = wid % 36;                 // 2304/64 pixel groups
  int rem = wid / 36;
  int tco = rem % 16;                 // 256/16 co tiles
  int b   = rem / 16;
  int half = lane >> 4, lm = lane & 15;
  // per-lane pixel -> padded-plane offset (tap (1,1) center => +1,+1 built in)
  int pofs[4];
#pragma unroll
  for (int g = 0; g < 4; ++g) {
    int s = tn * 64 + g * 16 + lm;
    int y = s / 48, x = s - y * 48;
    pofs[g] = y * kPW + x;            // tap adds ky*50 + kx
  }
  const float* Aw = w + (size_t)(tco * 16 + lm) * (kC * 9);  // + ci*9 + tap
  const float* Pb = pad + (size_t)b * kC * kPP;              // + ci*2500 + off
  v8f a0 = {}, a1 = {}, a2 = {}, a3 = {};
#pragma unroll
  for (int tap = 0; tap < 9; ++tap) {
    int ky = tap / 3, kx = tap - ky * 3;
    int toff = ky * kPW + kx;
    const float* B0 = Pb + pofs[0] + toff;
    const float* B1 = Pb + pofs[1] + toff;
    const float* B2 = Pb + pofs[2] + toff;
    const float* B3 = Pb + pofs[3] + toff;
    const float* At = Aw + tap;
    for (int k0 = 0; k0 < kC; k0 += 4) {
      int ka = k0 + 2 * half;
      v2f a;  a.x  = At[ka * 9];              a.y  = At[(ka + 1) * 9];
      v2f b0; b0.x = B0[(size_t)ka * kPP];    b0.y = B0[(size_t)(ka + 1) * kPP];
      v2f b1; b1.x = B1[(size_t)ka * kPP];    b1.y = B1[(size_t)(ka + 1) * kPP];
      v2f b2; b2.x = B2[(size_t)ka * kPP];    b2.y = B2[(size_t)(ka + 1) * kPP];
      v2f b3; b3.x = B3[(size_t)ka * kPP];    b3.y = B3[(size_t)(ka + 1) * kPP];
      a0 = wmma_f32(a, b0, a0);
      a1 = wmma_f32(a, b1, a1);
      a2 = wmma_f32(a, b2, a2);
      a3 = wmma_f32(a, b3, a3);
    }
  }
  const float kScale      = 1.0507009873554805f;
  const float kAlphaScale = 1.0507009873554805f * 1.6732632423543772f;
  v8f accs[4] = {a0, a1, a2, a3};
#pragma unroll
  for (int g = 0; g < 4; ++g) {
    int col = tn * 64 + g * 16 + lm;         // output pixel s
#pragma unroll
    for (int rr = 0; rr < 8; ++rr) {
      int row = tco * 16 + rr + 8 * half;    // output channel co
      float vv = accs[g][rr] + bias[row];
      if (mode) {
        vv = (vv > 0.0f) ? kScale * vv : kAlphaScale * expm1f(vv);
        int y = col / 48, x = col - y * 48;
        outPad[((size_t)(b * kC + row)) * kPP + (size_t)(y + 1) * kPW + (x + 1)] = vv;
      } else {
        outFlat[((size_t)(b * kC + row)) * kS + col] = vv;
      }
    }
  }
}

// ---------------------------------------------------------------------------
extern "C" void kernel_launch(void* const* d_in, const int* in_sizes, int n_in,
                              void* d_out, int out_size, void* d_ws, size_t ws_size,
                              hipStream_t stream) {
  (void)in_sizes; (void)n_in; (void)out_size; (void)ws_size;
  const float* x   = (const float*)d_in[0];
  const float* ref = (const float*)d_in[1];
  const float* qW1 = (const float*)d_in[2];  const float* qb1 = (const float*)d_in[3];
  const float* qW2 = (const float*)d_in[4];  const float* qb2 = (const float*)d_in[5];
  const float* kW1 = (const float*)d_in[6];  const float* kb1 = (const float*)d_in[7];
  const float* kW2 = (const float*)d_in[8];  const float* kb2 = (const float*)d_in[9];
  const float* vW1 = (const float*)d_in[10]; const float* vb1 = (const float*)d_in[11];
  const float* vW2 = (const float*)d_in[12]; const float* vb2 = (const float*)d_in[13];
  const float* c1W = (const float*)d_in[14]; const float* c1b = (const float*)d_in[15];
  const float* c2W = (const float*)d_in[16]; const float* c2b = (const float*)d_in[17];

  float* outp  = (float*)d_out;
  float* probs = outp + kNE;           // second tuple element: [B,H,S,S]
  float* ws = (float*)d_ws;
  float* content = ws + 0 * kNE;
  float* refer   = ws + 1 * kNE;
  float* hid     = ws + 2 * kNE;
  float* qv      = ws + 3 * kNE;
  float* kv      = ws + 4 * kNE;
  float* vv      = ws + 5 * kNE;
  float* pad1    = ws + 6 * kNE;       // [B,C,50,50] conv1 input (ctx output)
  float* pad2    = pad1 + kNP;         // [B,C,50,50] conv2 input (selu(conv1))
  // total ws use: 6*kNE + 2*kNP floats = ~38.6 MB

  zero_kernel<<<10000, 256, 0, stream>>>(pad1);           // zeros pad1 + pad2
  transpose_kernel<<<4608, 256, 0, stream>>>(x, ref, content, refer);

  // Deep embeds: M=4608, K=N=256 -> 288 m-tiles * 4 n-groups / 8 = 144 blocks.
  gemm_bias_act_kernel<<<144, 256, 0, stream>>>(content, qW1, qb1, hid, 256, 256, 1);
  gemm_bias_act_kernel<<<144, 256, 0, stream>>>(hid,     qW2, qb2, qv,  256, 256, 1);
  gemm_bias_act_kernel<<<144, 256, 0, stream>>>(refer,   kW1, kb1, hid, 256, 256, 1);
  gemm_bias_act_kernel<<<144, 256, 0, stream>>>(hid,     kW2, kb2, kv,  256, 256, 1);
  gemm_bias_act_kernel<<<144, 256, 0, stream>>>(refer,   vW1, vb1, hid, 256, 256, 1);
  gemm_bias_act_kernel<<<144, 256, 0, stream>>>(hid,     vW2, vb2, vv,  256, 256, 1);

  scores_kernel  <<<10368, 256, 0, stream>>>(qv, kv, probs); // 16*144*36 / 8
  headmask_kernel<<<41472, 256, 0, stream>>>(probs);         // 2*2304*2304/256
  softmax_kernel <<<36864, 256, 0, stream>>>(probs);         // 2*8*2304 rows
  ctx_kernel     <<<288,   256, 0, stream>>>(probs, vv, pad1); // 16*144 / 8

  // Convs as WMMA: 2 * 16 co-tiles * 36 pix-groups / 8 = 144 blocks each.
  conv3x3_wmma_kernel<<<144, 256, 0, stream>>>(pad1, c1W, c1b, pad2, nullptr, 1);
  conv3x3_wmma_kernel<<<144, 256, 0, stream>>>(pad2, c2W, c2b, nullptr, outp, 0);
}